// BSNN_70789650973458
// MI455X (gfx1250) — compile-verified
//
#include <hip/hip_runtime.h>

typedef float v2f __attribute__((ext_vector_type(2)));
typedef float v8f __attribute__((ext_vector_type(8)));

#define N_ROWS      131072
#define ROWS_PER_WG 64
#define STR         132   // padded LDS row stride (128+4) -> conflict-free

__device__ __forceinline__ v8f bias_init(float bv) {
    v8f c;
#pragma unroll
    for (int i = 0; i < 8; ++i) c[i] = bv;
    return c;
}

// One 64(K) x 16(N) strip of D = A @ B + C for all FOUR 16-row tiles of a
// 64-row slab, sharing each B fragment across 4 WMMAs.  A comes from LDS
// (row-major, stride STR, starting column srcColOff), B from global weights W
// (row-major, leading dim ldW; element W[(kOff+k)*ldW + nCol]).
// Fragment layouts per CDNA5 ISA 7.12.2 (32-bit A 16x4: lanes0-15 K={0,1},
// lanes16-31 K={2,3}; B 4x16 mirrored; C/D: VGPR v = rows v / v+8).
__device__ __forceinline__ void wmma_k64(const float* __restrict__ W, int ldW,
                                         int kOff, int nCol,
                                         const float* src, int srcColOff,
                                         int lane, v8f c[4]) {
    const int half = lane >> 4;
    const int l15  = lane & 15;
#pragma unroll
    for (int kk = 0; kk < 16; ++kk) {
        const int k0 = kk * 4 + 2 * half;
        v2f b;
        b.x = W[(kOff + k0 + 0) * ldW + nCol];
        b.y = W[(kOff + k0 + 1) * ldW + nCol];
#pragma unroll
        for (int rt = 0; rt < 4; ++rt) {
            v2f a = *(const v2f*)&src[(16 * rt + l15) * STR + srcColOff + k0];
            c[rt] = __builtin_amdgcn_wmma_f32_16x16x4_f32(
                false, a, false, b, (short)0, c[rt], false, false);
        }
    }
}

__device__ __forceinline__ void sin_store4(const v8f c[4], float* dst,
                                           int nCol, int half) {
#pragma unroll
    for (int rt = 0; rt < 4; ++rt)
#pragma unroll
        for (int v = 0; v < 8; ++v)
            dst[(16 * rt + v + 8 * half) * STR + nCol] = __sinf(c[rt][v]);
}

__global__ __launch_bounds__(256)
void bsnn_fused(const float* __restrict__ X,
                const float* __restrict__ W0, const float* __restrict__ B0,
                const float* __restrict__ W1, const float* __restrict__ B1,
                const float* __restrict__ W2, const float* __restrict__ B2,
                const float* __restrict__ W3, const float* __restrict__ B3,
                const float* __restrict__ W4, const float* __restrict__ B4,
                const float* __restrict__ W5, const float* __restrict__ B5,
                float* __restrict__ out) {
    __shared__ float hA[ROWS_PER_WG * STR];  // 64x128 h1
    __shared__ float hB[ROWS_PER_WG * STR];  // 64x128 current h2 chunk
    __shared__ float hC[ROWS_PER_WG * STR];  // h0 (64x64), later h3 chunks
    __shared__ float acc[ROWS_PER_WG];       // per-row output accumulator

    const int tid      = threadIdx.x;
    const int w        = tid >> 5;   // wave id 0..7
    const int lane     = tid & 31;
    const int half     = lane >> 4;
    const int l15      = lane & 15;
    const int rowsBase = blockIdx.x * ROWS_PER_WG;
    const int nCol     = 16 * w + l15;  // this wave's N-strip column

    if (tid < ROWS_PER_WG) acc[tid] = 0.0f;

    // -------- layer 0: X(64x3) @ W0(3x64), K padded to 4; h0 -> hC ---------
    // Branch-free: load all three X elements / W0 rows in every lane, then
    // select per half with v_cndmask (keeps EXEC all-ones, no saveexec).
    {
        const int nc0 = 16 * (w & 3) + l15;  // 4 N-tiles over 64-wide output
        const float bias = B0[nc0];
        const float w0r = W0[0 * 64 + nc0];
        const float w1r = W0[1 * 64 + nc0];
        const float w2r = W0[2 * 64 + nc0];
        v2f b;
        b.x = half ? w2r : w0r;
        b.y = half ? 0.0f : w1r;
#pragma unroll
        for (int j = 0; j < 2; ++j) {          // 2 of the 4 row tiles per wave
            const int rB   = 16 * (w >> 2) + 32 * j;
            const int gRow = rowsBase + rB + l15;
            const float x0 = X[gRow * 3 + 0];
            const float x1 = X[gRow * 3 + 1];
            const float x2 = X[gRow * 3 + 2];
            v2f a;
            a.x = half ? x2 : x0;
            a.y = half ? 0.0f : x1;
            v8f c = bias_init(bias);
            c = __builtin_amdgcn_wmma_f32_16x16x4_f32(
                false, a, false, b, (short)0, c, false, false);
#pragma unroll
            for (int v = 0; v < 8; ++v)
                hC[(rB + v + 8 * half) * STR + nc0] = __sinf(c[v]);
        }
    }
    __syncthreads();

    // -------- layer 1: h0(64x64) @ W1(64x128) -> hA ------------------------
    {
        v8f c[4];
#pragma unroll
        for (int rt = 0; rt < 4; ++rt) c[rt] = bias_init(B1[nCol]);
        wmma_k64(W1, 128, 0, nCol, hC, 0, lane, c);
        sin_store4(c, hA, nCol, half);
    }
    __syncthreads();

    // Per-lane output partials, accumulated across all 8 h4 chunks; reduced
    // across lanes only once at the very end.
    v8f pacc[4];
#pragma unroll
    for (int rt = 0; rt < 4; ++rt) pacc[rt] = bias_init(0.0f);

    // -------- tree walk over block-diagonal layers 2..4 + output -----------
    for (int c2 = 0; c2 < 2; ++c2) {
        {   // layer 2 chunk c2: hA[:,64c2..] @ W2 block -> hB
            v8f c[4];
#pragma unroll
            for (int rt = 0; rt < 4; ++rt)
                c[rt] = bias_init(B2[128 * c2 + nCol]);
            wmma_k64(W2, 256, 64 * c2, 128 * c2 + nCol, hA, 64 * c2, lane, c);
            sin_store4(c, hB, nCol, half);
        }
        __syncthreads();
        for (int c3 = 0; c3 < 2; ++c3) {
            const int i3 = 2 * c2 + c3;
            {   // layer 3 chunk i3: hB[:,64c3..] @ W3 block -> hC
                v8f c[4];
#pragma unroll
                for (int rt = 0; rt < 4; ++rt)
                    c[rt] = bias_init(B3[128 * i3 + nCol]);
                wmma_k64(W3, 512, 64 * i3, 128 * i3 + nCol, hB, 64 * c3,
                         lane, c);
                sin_store4(c, hC, nCol, half);
            }
            __syncthreads();
            for (int c4 = 0; c4 < 2; ++c4) {
                const int i4 = 2 * i3 + c4;
                // layer 4 chunk i4 (registers only) fused with output layer
                v8f c[4];
#pragma unroll
                for (int rt = 0; rt < 4; ++rt)
                    c[rt] = bias_init(B4[128 * i4 + nCol]);
                wmma_k64(W4, 1024, 64 * i4, 128 * i4 + nCol, hC, 64 * c4,
                         lane, c);
                const float w5v = W5[128 * i4 + nCol];
#pragma unroll
                for (int rt = 0; rt < 4; ++rt)
#pragma unroll
                    for (int v = 0; v < 8; ++v)
                        pacc[rt][v] += __sinf(c[rt][v]) * w5v;
            }
            __syncthreads();  // hC consumers done before next c3 rewrites hC
        }
    }

    // -------- final cross-lane reduction of output partials ----------------
#pragma unroll
    for (int rt = 0; rt < 4; ++rt) {
#pragma unroll
        for (int v = 0; v < 8; ++v) {
            float p = pacc[rt][v];
            p += __shfl_xor(p, 1, 16);
            p += __shfl_xor(p, 2, 16);
            p += __shfl_xor(p, 4, 16);
            p += __shfl_xor(p, 8, 16);
            if (l15 == 0) atomicAdd(&acc[16 * rt + v + 8 * half], p);
        }
    }
    __syncthreads();
    if (tid < ROWS_PER_WG) out[rowsBase + tid] = acc[tid] + B5[0];
}

extern "C" void kernel_launch(void* const* d_in, const int* in_sizes, int n_in,
                              void* d_out, int out_size, void* d_ws,
                              size_t ws_size, hipStream_t stream) {
    (void)in_sizes; (void)n_in; (void)d_ws; (void)ws_size; (void)out_size;
    const float* X  = (const float*)d_in[0];
    const float* W0 = (const float*)d_in[1];
    const float* B0 = (const float*)d_in[2];
    const float* W1 = (const float*)d_in[3];
    const float* B1 = (const float*)d_in[4];
    const float* W2 = (const float*)d_in[5];
    const float* B2 = (const float*)d_in[6];
    const float* W3 = (const float*)d_in[7];
    const float* B3 = (const float*)d_in[8];
    const float* W4 = (const float*)d_in[9];
    const float* B4 = (const float*)d_in[10];
    const float* W5 = (const float*)d_in[11];
    const float* B5 = (const float*)d_in[12];

    dim3 grid(N_ROWS / ROWS_PER_WG);   // 2048 workgroups
    dim3 block(256);                   // 8 wave32
    bsnn_fused<<<grid, block, 0, stream>>>(X, W0, B0, W1, B1, W2, B2, W3, B3,
                                           W4, B4, W5, B5, (float*)d_out);
}